// SoftHardQuantize_71760313582210
// MI455X (gfx1250) — compile-verified
//
#include <hip/hip_runtime.h>
#include <math.h>

typedef __attribute__((ext_vector_type(2))) float v2f;
typedef __attribute__((ext_vector_type(8))) float v8f;

#define D 64
#define K 512
#define NROWS (16 * 16384)          // 262144 rows
#define ET_STRIDE 66                 // padded (even -> 8B-aligned float2 LDS loads)
#define SC_STRIDE 516                // padded logit scratch stride (conflict-free halves)
#define WAVES 4
#define BLOCK (WAVES * 32)
#define GRID 512
#define N_TILES64 (NROWS / 64)       // 4096

// ---------------------------------------------------------------------------
// zero the global likelihood accumulator in d_ws (512 floats)
// ---------------------------------------------------------------------------
extern "C" __global__ void vq_init(float* __restrict__ gacc) {
    int i = blockIdx.x * blockDim.x + threadIdx.x;
    if (i < K) gacc[i] = 0.0f;
}

// ---------------------------------------------------------------------------
// main kernel: fp32 WMMA GEMM -> softmax stats -> argmax gather
// LDS layout (dynamic):
//   s_embedT : K * ET_STRIDE floats   (embed transposed, [code][dim], padded)
//   s_e2     : K floats               (||e_k||^2)
//   s_lik    : K floats               (block-local likelihood accumulator)
//   s_scr    : WAVES * 16*SC_STRIDE   (per-wave logit/exp scratch)
//   s_idx    : WAVES * 16 ints        (per-wave argmax indices)
// total ~271.6 KB of the 320 KB/WGP budget -> 1 WG per WGP, 1 wave per SIMD.
// ---------------------------------------------------------------------------
extern "C" __global__ void vq_main(const float* __restrict__ x,
                                   const float* __restrict__ embed,
                                   const float* __restrict__ sigma_p,
                                   float* __restrict__ qout,
                                   float* __restrict__ gacc) {
    extern __shared__ float smem[];
    float* s_embedT = smem;                              // K*ET_STRIDE
    float* s_e2     = s_embedT + K * ET_STRIDE;          // K
    float* s_lik    = s_e2 + K;                          // K
    float* s_scr    = s_lik + K;                         // WAVES*16*SC_STRIDE
    int*   s_idx    = (int*)(s_scr + WAVES * 16 * SC_STRIDE);

    const int tid  = threadIdx.x;
    const int lane = tid & 31;
    const int wave = tid >> 5;
    const int half = lane >> 4;      // 0: lanes 0-15, 1: lanes 16-31
    const int l16  = lane & 15;
    const float sigma = sigma_p[0];

    // ---- stage embed^T into LDS (coalesced global read, transposed write) ----
    for (int i = tid; i < D * K; i += BLOCK) {
        int d = i >> 9;              // i / 512
        int n = i & (K - 1);         // i % 512
        s_embedT[n * ET_STRIDE + d] = embed[i];
    }
    for (int i = tid; i < K; i += BLOCK) s_lik[i] = 0.0f;
    __syncthreads();
    for (int n = tid; n < K; n += BLOCK) {
        const float* row = &s_embedT[n * ET_STRIDE];
        float s = 0.0f;
#pragma unroll
        for (int d = 0; d < D; ++d) s += row[d] * row[d];
        s_e2[n] = s;
    }
    __syncthreads();

    float* scr = s_scr + wave * (16 * SC_STRIDE);

    for (int it = blockIdx.x; it < N_TILES64; it += gridDim.x) {
        const int rowbase = it * 64 + wave * 16;

        // ---- A fragments: 16x4 f32 per k-step; lane half selects K pair ----
        // lane l16 holds row M=l16; VGPR pair holds (d0, d0+1), d0 = 4*kk + 2*half
        v2f a[16];
        const float* xrow = x + (size_t)(rowbase + l16) * D + 2 * half;
#pragma unroll
        for (int kk = 0; kk < 16; ++kk)
            a[kk] = *(const v2f*)(xrow + kk * 4);

        float rmax[8]; int ridx[8];
#pragma unroll
        for (int j = 0; j < 8; ++j) { rmax[j] = -3.4e38f; ridx[j] = 0; }

        // ---- pass 1: dual-tile WMMA (two independent accumulation chains
        //      sharing the A fragments), stage logits, track row max/argmax ----
        for (int c = 0; c < K / 16; c += 2) {
            const int n0 = c * 16 + l16;             // code column, tile c
            const int n1 = n0 + 16;                  // code column, tile c+1
            const float* brow0 = &s_embedT[n0 * ET_STRIDE];
            const float* brow1 = &s_embedT[n1 * ET_STRIDE];
            v8f acc0 = {};
            v8f acc1 = {};
#pragma unroll
            for (int kk = 0; kk < 16; ++kk) {
                const int d0 = kk * 4 + 2 * half;
                v2f b0 = *(const v2f*)(brow0 + d0);   // ds_load_b64
                v2f b1 = *(const v2f*)(brow1 + d0);
                acc0 = __builtin_amdgcn_wmma_f32_16x16x4_f32(
                    false, a[kk], false, b0, (short)0, acc0, false, false);
                acc1 = __builtin_amdgcn_wmma_f32_16x16x4_f32(
                    false, a[kk], false, b1, (short)0, acc1, false, false);
            }
            const float e20 = s_e2[n0];
            const float e21 = s_e2[n1];
#pragma unroll
            for (int j = 0; j < 8; ++j) {
                // shift-invariant logit: t = sigma * (2*x.e - ||e||^2)
                float t0 = sigma * (2.0f * acc0[j] - e20);
                float t1 = sigma * (2.0f * acc1[j] - e21);
                scr[(j + 8 * half) * SC_STRIDE + n0] = t0;
                scr[(j + 8 * half) * SC_STRIDE + n1] = t1;
                if (t0 > rmax[j]) { rmax[j] = t0; ridx[j] = n0; }
                if (t1 > rmax[j]) { rmax[j] = t1; ridx[j] = n1; }
            }
        }

        // ---- reduce max+argmax across the 16 lanes holding each row ----
#pragma unroll
        for (int j = 0; j < 8; ++j) {
            float m = rmax[j]; int mi = ridx[j];
#pragma unroll
            for (int s = 8; s >= 1; s >>= 1) {
                float om = __shfl_xor(m, s, 32);
                int   oi = __shfl_xor(mi, s, 32);
                if (om > m || (om == m && oi < mi)) { m = om; mi = oi; }
            }
            rmax[j] = m; ridx[j] = mi;
        }
        if (l16 == 0) {
#pragma unroll
            for (int j = 0; j < 8; ++j) s_idx[wave * 16 + 8 * half + j] = ridx[j];
        }

        // ---- pass 2: exp + row partition function ----
        float zp[8];
#pragma unroll
        for (int j = 0; j < 8; ++j) zp[j] = 0.0f;
        for (int c = 0; c < K / 16; ++c) {
            const int n = c * 16 + l16;
#pragma unroll
            for (int j = 0; j < 8; ++j) {
                float* p = &scr[(j + 8 * half) * SC_STRIDE + n];
                float e = __expf(*p - rmax[j]);
                *p = e;
                zp[j] += e;
            }
        }
#pragma unroll
        for (int j = 0; j < 8; ++j) {
            float z = zp[j];
#pragma unroll
            for (int s = 8; s >= 1; s >>= 1) z += __shfl_xor(z, s, 32);
            zp[j] = 1.0f / z;                         // invZ, uniform in half
        }

        // ---- pass 3: likelihood partials -> LDS atomic accumulator ----
        for (int c = 0; c < K / 16; ++c) {
            const int n = c * 16 + l16;
            float s = 0.0f;
#pragma unroll
            for (int j = 0; j < 8; ++j)
                s += scr[(j + 8 * half) * SC_STRIDE + n] * zp[j];
            atomicAdd(&s_lik[n], s);                  // ds_add_f32
        }

        // ---- hard-gather output: forward(quantize) == embed[:, argmax].T ----
        for (int r = 0; r < 16; ++r) {
            int idx = s_idx[wave * 16 + r];
            v2f hv = *(const v2f*)&s_embedT[idx * ET_STRIDE + 2 * lane];
            *(v2f*)(qout + (size_t)(rowbase + r) * D + 2 * lane) = hv;
        }
    }

    // ---- flush block-local likelihood sums ----
    __syncthreads();
    for (int n = tid; n < K; n += BLOCK)
        atomicAdd(&gacc[n], s_lik[n]);
}

// ---------------------------------------------------------------------------
// finalize: likelihoods = acc/N ; quant_loss = C * mean_k p*(log p - log(l+eps))
// ---------------------------------------------------------------------------
extern "C" __global__ void vq_final(const float* __restrict__ gacc,
                                    float* __restrict__ out) {
    __shared__ float red[K];
    const int k = threadIdx.x;
    const float lik = gacc[k] * (1.0f / (float)NROWS);
    out[(size_t)NROWS * D + 1 + k] = lik;
    const float p = 1.0f / (float)K;
    red[k] = p * (logf(p) - logf(lik + 1e-10f));
    __syncthreads();
    for (int s = K / 2; s > 0; s >>= 1) {
        if (k < s) red[k] += red[k + s];
        __syncthreads();
    }
    if (k == 0) out[(size_t)NROWS * D] = 0.25f * red[0] * (1.0f / (float)K);
}

// ---------------------------------------------------------------------------
extern "C" void kernel_launch(void* const* d_in, const int* in_sizes, int n_in,
                              void* d_out, int out_size, void* d_ws, size_t ws_size,
                              hipStream_t stream) {
    const float* x     = (const float*)d_in[0];
    const float* embed = (const float*)d_in[1];
    const float* sigma = (const float*)d_in[2];
    float* out  = (float*)d_out;
    float* gacc = (float*)d_ws;   // 512-float likelihood accumulator

    const size_t shmem = (size_t)(K * ET_STRIDE + K + K + WAVES * 16 * SC_STRIDE) * sizeof(float)
                       + (size_t)(WAVES * 16) * sizeof(int);
    (void)hipFuncSetAttribute(reinterpret_cast<const void*>(&vq_main),
                              hipFuncAttributeMaxDynamicSharedMemorySize, (int)shmem);

    hipLaunchKernelGGL(vq_init,  dim3(2),    dim3(256),  0,     stream, gacc);
    hipLaunchKernelGGL(vq_main,  dim3(GRID), dim3(BLOCK), shmem, stream, x, embed, sigma, out, gacc);
    hipLaunchKernelGGL(vq_final, dim3(1),    dim3(K),    0,     stream, gacc, out);
}